// clDiceLoss_48765058679424
// MI455X (gfx1250) — compile-verified
//
#include <hip/hip_runtime.h>
#include <math.h>

#define N 128
#define VOL (128*128*128)
#define TILE 16
#define S 20                 // TILE + 2*halo
#define SMOOTH 1e-6f
#define INV3 (1.0f/3.0f)
#define NUM_ITER 40

typedef __attribute__((ext_vector_type(2))) float v2f;
typedef __attribute__((ext_vector_type(8))) float v8f;

// LDS cube index, coords in [-2, 18)
__device__ __forceinline__ int IDX(int x, int y, int z) {
    return ((x + 2) * S + (y + 2)) * S + (z + 2);
}
__device__ __forceinline__ float clamp01(float v) {
    return v < 0.f ? 0.f : (v > 1.f ? 1.f : v);
}

// ---------------- init / prep ----------------

__global__ void init_kernel(int* flags, int* use_sig) {
    if (threadIdx.x == 0) { flags[0] = 0; flags[1] = 0; *use_sig = 0; }
}

__global__ void max_kernel(const float* __restrict__ pred, int* use_sig) {
    size_t stride = (size_t)gridDim.x * blockDim.x;
    for (size_t i = (size_t)blockIdx.x * blockDim.x + threadIdx.x; i < 2u * VOL; i += stride)
        if (pred[i] > 1.0f) *use_sig = 1;   // idempotent store, race-free by value
}

__global__ void prep_kernel(const float* __restrict__ pred, const float* __restrict__ targ,
                            float* __restrict__ cur, const int* __restrict__ use_sig) {
    int sig = *use_sig;
    size_t stride = (size_t)gridDim.x * blockDim.x;
    for (size_t i = (size_t)blockIdx.x * blockDim.x + threadIdx.x; i < 4u * VOL; i += stride) {
        int tb = (int)(i / VOL);            // 0..3 : (tensor<<1)|batch
        size_t e = i - (size_t)tb * VOL;
        int t = tb >> 1, b = tb & 1;
        float v;
        if (t == 0) {
            v = pred[(size_t)b * VOL + e];
            if (sig) v = 1.f / (1.f + expf(-v));
        } else {
            v = targ[(size_t)b * VOL + e];
        }
        cur[i] = clamp01(v);
    }
}

// ---------------- fused skeletonization step ----------------
// grid: (512 tiles, 2 batch, 2 tensor), block: 256, dyn LDS: 3*8000 floats

__global__ __launch_bounds__(256) void step_kernel(const float* __restrict__ cur,
                                                   float* __restrict__ nxt,
                                                   const int* __restrict__ flags,
                                                   float* __restrict__ partials) {
    const int t = blockIdx.z, b = blockIdx.y;
    const int tile = blockIdx.x;
    const int gx0 = (tile >> 6) * TILE;
    const int gy0 = ((tile >> 3) & 7) * TILE;
    const int gz0 = (tile & 7) * TILE;
    const float* src = cur + (size_t)(t * 2 + b) * VOL;
    float* dst = nxt + (size_t)(t * 2 + b) * VOL;
    const int tid = threadIdx.x;

    if (flags[t]) {                          // frozen: uniform copy path
        for (int i = tid; i < 4096; i += 256) {
            int lx = i >> 8, ly = (i >> 4) & 15, lz = i & 15;
            size_t g = ((size_t)(gx0 + lx) * N + (gy0 + ly)) * N + (gz0 + lz);
            dst[g] = src[g];
        }
        return;
    }

    extern __shared__ float smem[];
    float* bufIn = smem;                     // 8000: input with halo 2
    float* bufB  = smem + 8000;
    float* bufC  = smem + 16000;

    // load 20^3 region (zero padded at volume edges)
    for (int i = tid; i < 8000; i += 256) {
        int lx = i / 400, r = i - lx * 400;
        int ly = r / S, lz = r - ly * S;
        int gx = gx0 + lx - 2, gy = gy0 + ly - 2, gz = gz0 + lz - 2;
        float v = 0.f;
        if ((unsigned)gx < (unsigned)N && (unsigned)gy < (unsigned)N && (unsigned)gz < (unsigned)N)
            v = src[((size_t)gx * N + gy) * N + gz];
        bufIn[i] = v;
    }
    __syncthreads();

    // conv1: Cz  (x,y in [-2,18), z in [-1,17))  -> bufB
    for (int i = tid; i < 20 * 20 * 18; i += 256) {
        int lx = i / 360, r = i - lx * 360;
        int ly = r / 18, lz = r - ly * 18;
        int x = lx - 2, y = ly - 2, z = lz - 1;
        bufB[IDX(x, y, z)] =
            (bufIn[IDX(x, y, z - 1)] + bufIn[IDX(x, y, z)] + bufIn[IDX(x, y, z + 1)]) * INV3;
    }
    __syncthreads();

    // conv1: Cy  (x in [-2,18), y,z in [-1,17)) -> bufC
    for (int i = tid; i < 20 * 18 * 18; i += 256) {
        int lx = i / 324, r = i - lx * 324;
        int ly = r / 18, lz = r - ly * 18;
        int x = lx - 2, y = ly - 1, z = lz - 1;
        bufC[IDX(x, y, z)] =
            (bufB[IDX(x, y - 1, z)] + bufB[IDX(x, y, z)] + bufB[IDX(x, y + 1, z)]) * INV3;
    }
    __syncthreads();

    // conv1: Cx + eroded = clip(. - s)  (x,y,z in [-1,17)) -> bufB
    for (int i = tid; i < 18 * 18 * 18; i += 256) {
        int lx = i / 324, r = i - lx * 324;
        int ly = r / 18, lz = r - ly * 18;
        int x = lx - 1, y = ly - 1, z = lz - 1;
        float v = (bufC[IDX(x - 1, y, z)] + bufC[IDX(x, y, z)] + bufC[IDX(x + 1, y, z)]) * INV3;
        bufB[IDX(x, y, z)] = clamp01(v - SMOOTH);
    }
    __syncthreads();

    // conv2: Cz  (x,y in [-1,17), z in [0,16)) -> bufC
    for (int i = tid; i < 18 * 18 * 16; i += 256) {
        int lx = i / 288, r = i - lx * 288;
        int ly = r / 16, lz = r - ly * 16;
        int x = lx - 1, y = ly - 1, z = lz;
        bufC[IDX(x, y, z)] =
            (bufB[IDX(x, y, z - 1)] + bufB[IDX(x, y, z)] + bufB[IDX(x, y, z + 1)]) * INV3;
    }
    __syncthreads();

    // conv2: Cy  (x in [-1,17), y,z in [0,16)) -> bufB ; zero x=-2,17 rows (WMMA zero-weight lanes)
    for (int i = tid; i < 18 * 16 * 16; i += 256) {
        int lx = i / 256, r = i - lx * 256;
        int ly = r / 16, lz = r - ly * 16;
        int x = lx - 1, y = ly, z = lz;
        bufB[IDX(x, y, z)] =
            (bufC[IDX(x, y - 1, z)] + bufC[IDX(x, y, z)] + bufC[IDX(x, y + 1, z)]) * INV3;
    }
    for (int i = tid; i < 512; i += 256) {
        int e = i >> 8, y = (i >> 4) & 15, z = i & 15;
        bufB[IDX(e ? 17 : -2, y, z)] = 0.f;
    }
    __syncthreads();

    // conv2: Cx via WMMA f32 16x16x4: opened(16x16) = T'(16x20) x t5(20x16) per z-slice.
    // T'[m][k] = 1/3 iff |(k-2)-m| <= 1 ; halo columns carried in K, edges have zero weight.
    const int lane  = tid & 31;
    const int wave  = tid >> 5;
    const int mrow  = lane & 15;        // A: M row / B: N col / D: N col
    const int khalf = lane >> 4;        // K half select per ISA 16x4/4x16 layout

    v2f aop[5];
    #pragma unroll
    for (int s5 = 0; s5 < 5; ++s5) {
        #pragma unroll
        for (int j = 0; j < 2; ++j) {
            int k  = 4 * s5 + j + 2 * khalf;
            int xk = k - 2;
            aop[s5][j] = (xk >= mrow - 1 && xk <= mrow + 1) ? INV3 : 0.f;
        }
    }

    float lsum = 0.f;
    #pragma unroll
    for (int sl = 0; sl < 2; ++sl) {
        int z = wave * 2 + sl;          // 8 waves x 2 slices = 16 z
        v8f acc = {0.f, 0.f, 0.f, 0.f, 0.f, 0.f, 0.f, 0.f};
        #pragma unroll
        for (int s5 = 0; s5 < 5; ++s5) {
            v2f bop;
            #pragma unroll
            for (int j = 0; j < 2; ++j) {
                int k = 4 * s5 + j + 2 * khalf;
                bop[j] = bufB[IDX(k - 2, mrow, z)];
            }
            acc = __builtin_amdgcn_wmma_f32_16x16x4_f32(false, aop[s5], false, bop,
                                                        0, acc, false, false);
        }
        // D layout: vgpr i, lanes 0-15 -> M=i, lanes 16-31 -> M=i+8 ; N = lane&15
        #pragma unroll
        for (int i = 0; i < 8; ++i) {
            int x = i + 8 * khalf, y = mrow;
            float op = clamp01(acc[i]);
            float nv = bufIn[IDX(x, y, z)] * (1.f - op + SMOOTH);
            dst[((size_t)(gx0 + x) * N + (gy0 + y)) * N + (gz0 + z)] = nv;
            lsum += nv;
        }
    }

    // deterministic block reduction of new.sum() contribution (bufC is dead)
    bufC[tid] = lsum;
    __syncthreads();
    for (int o = 128; o > 0; o >>= 1) {
        if (tid < o) bufC[tid] += bufC[tid + o];
        __syncthreads();
    }
    if (tid == 0) partials[((t * 2 + b) << 9) + tile] = bufC[0];
}

__global__ void flag_kernel(int* flags, const float* __restrict__ partials) {
    if (threadIdx.x < 2) {
        int t = threadIdx.x;
        float s = 0.f;
        for (int i = 0; i < 1024; ++i) s += partials[t * 1024 + i];  // fixed order
        if (s < SMOOTH) flags[t] = 1;                                 // sticky
    }
}

// ---------------- centerline dice ----------------

__global__ __launch_bounds__(256) void dice_kernel(const float* __restrict__ skel,
                                                   float* __restrict__ dpart) {
    int b = blockIdx.y;
    const float* p  = skel + (size_t)(0 * 2 + b) * VOL;
    const float* tg = skel + (size_t)(1 * 2 + b) * VOL;
    float si = 0.f, sp = 0.f, st = 0.f;
    size_t stride = (size_t)gridDim.x * blockDim.x;
    for (size_t i = (size_t)blockIdx.x * blockDim.x + threadIdx.x; i < VOL; i += stride) {
        float pv = p[i], tv = tg[i];
        si += pv * tv; sp += pv; st += tv;
    }
    __shared__ float sh[3][256];
    int tid = threadIdx.x;
    sh[0][tid] = si; sh[1][tid] = sp; sh[2][tid] = st;
    __syncthreads();
    for (int o = 128; o > 0; o >>= 1) {
        if (tid < o) {
            sh[0][tid] += sh[0][tid + o];
            sh[1][tid] += sh[1][tid + o];
            sh[2][tid] += sh[2][tid + o];
        }
        __syncthreads();
    }
    if (tid == 0) {
        size_t o = ((size_t)b * 256 + blockIdx.x) * 3;
        dpart[o + 0] = sh[0][0]; dpart[o + 1] = sh[1][0]; dpart[o + 2] = sh[2][0];
    }
}

__global__ void final_kernel(const float* __restrict__ dpart, float* __restrict__ out) {
    if (threadIdx.x == 0) {
        float loss = 0.f;
        for (int b = 0; b < 2; ++b) {
            float I = 0.f, P = 0.f, T = 0.f;
            for (int i = 0; i < 256; ++i) {          // fixed order
                size_t o = ((size_t)b * 256 + i) * 3;
                I += dpart[o + 0]; P += dpart[o + 1]; T += dpart[o + 2];
            }
            float dice = (2.f * I + SMOOTH) / (P + T + SMOOTH);
            loss += (1.f - dice);
        }
        out[0] = loss * 0.5f;
    }
}

// ---------------- host ----------------

extern "C" void kernel_launch(void* const* d_in, const int* in_sizes, int n_in,
                              void* d_out, int out_size, void* d_ws, size_t ws_size,
                              hipStream_t stream) {
    const float* pred = (const float*)d_in[0];
    const float* targ = (const float*)d_in[1];
    float* out = (float*)d_out;

    float* buf0 = (float*)d_ws;                        // [2 tensors][2 batch][VOL]
    float* buf1 = buf0 + (size_t)4 * VOL;
    float* tail = buf1 + (size_t)4 * VOL;
    int* flags    = (int*)tail;                        // 2
    int* use_sig  = flags + 2;                         // 1 (+1 pad)
    float* stepPart = (float*)(use_sig + 2);           // 2048
    float* dicePart = stepPart + 2048;                 // 2*256*3

    init_kernel<<<1, 32, 0, stream>>>(flags, use_sig);
    max_kernel<<<2048, 256, 0, stream>>>(pred, use_sig);
    prep_kernel<<<4096, 256, 0, stream>>>(pred, targ, buf0, use_sig);

    dim3 sgrid(512, 2, 2);
    const size_t lds = 3 * 8000 * sizeof(float);       // 96 KB dynamic LDS
    for (int it = 0; it < NUM_ITER; ++it) {
        float* cur = (it & 1) ? buf1 : buf0;
        float* nxt = (it & 1) ? buf0 : buf1;
        step_kernel<<<sgrid, 256, lds, stream>>>(cur, nxt, flags, stepPart);
        flag_kernel<<<1, 32, 0, stream>>>(flags, stepPart);
    }
    // NUM_ITER even -> final skeletons in buf0

    dice_kernel<<<dim3(256, 2), 256, 0, stream>>>(buf0, dicePart);
    final_kernel<<<1, 32, 0, stream>>>(dicePart, out);
}